// Social_Aggregator_42906723287403
// MI455X (gfx1250) — compile-verified
//
#include <hip/hip_runtime.h>

#define N_NODES 16384
#define DEGREE 32
#define EMBED 64
#define WAVES_PER_BLOCK 4
#define BLOCK_THREADS (WAVES_PER_BLOCK * 32)

// LDS pitch (floats) for paired-row weight layout: element (k,n) lives at
// [(k>>1)*P_PITCH + 2n + (k&1)]. Pair-row advance = 160 dwords = 32 banks, so
// half-wave 0 (pair-row pr) uses banks [0,32) and half-wave 1 (pr+1) uses
// banks [32,64): conflict-free b64 B-fragment loads with zero repacking movs.
#define P_PITCH 160
#define R_PITCH 68 // nbr/h1 rows: 4-bank advance/row -> A-frag b64 reads conflict-free

typedef __attribute__((ext_vector_type(2))) float v2f;
typedef __attribute__((ext_vector_type(8))) float v8f;

// LDS layout (in floats)
#define LW1_F (64 * P_PITCH) // 64 pair-rows (K=128)
#define LW2_F (32 * P_PITCH) // 32 pair-rows (K=64)
#define NODE_F (DEGREE * R_PITCH /*nbr*/ + DEGREE * R_PITCH /*h1*/ + EMBED /*u*/ + DEGREE /*att*/)
#define SMEM_F (LW1_F + LW2_F + 64 /*W3*/ + 64 /*b1*/ + 64 /*b2*/ + WAVES_PER_BLOCK * NODE_F)

static __device__ __forceinline__ void lds_fence() {
  // same-wave LDS RAW ordering + compiler reorder barrier
  asm volatile("s_wait_dscnt 0" ::: "memory");
}

static __device__ __forceinline__ v8f wmma_f32(v2f a, v2f b, v8f c) {
  // D = A(16x4 f32) * B(4x16 f32) + C(16x16 f32)
  return __builtin_amdgcn_wmma_f32_16x16x4_f32(false, a, false, b, (short)0, c, false, false);
}

__global__ __launch_bounds__(BLOCK_THREADS) void social_agg_kernel(
    const int* __restrict__ nodes,
    const int* __restrict__ nbr_idx,
    const float* __restrict__ emb,
    const float* __restrict__ W1,
    const float* __restrict__ b1,
    const float* __restrict__ W2,
    const float* __restrict__ b2,
    const float* __restrict__ W3,
    float* __restrict__ out) {
  extern __shared__ float smem[];
  float* lW1 = smem;          // paired-row W1
  float* lW2 = lW1 + LW1_F;   // paired-row W2
  float* lW3 = lW2 + LW2_F;
  float* lb1 = lW3 + 64;
  float* lb2 = lb1 + 64;
  float* nodeBase = lb2 + 64;

  const int tid = threadIdx.x;
  const int wave = tid >> 5;
  const int lane = tid & 31;
  const int hi = lane >> 4; // half-wave (K-pair / M-group selector)
  const int lo = lane & 15; // row (A) or column (B/C/D) within tile

  // ---- stage weights into LDS in paired-row layout (whole block, once) ----
  {
    // W1: 64 pair-rows x 32 column-pairs; (k,n) -> lW1[(k>>1)*P_PITCH + 2n + (k&1)]
#pragma unroll 4
    for (int i = tid; i < 64 * 32; i += BLOCK_THREADS) {
      int pr = i >> 5, n2 = (i & 31) << 1;
      float2 ra = *(const float2*)&W1[(2 * pr + 0) * 64 + n2];
      float2 rb = *(const float2*)&W1[(2 * pr + 1) * 64 + n2];
      *(float4*)&lW1[pr * P_PITCH + 2 * n2] = make_float4(ra.x, rb.x, ra.y, rb.y);
    }
    // W2: 32 pair-rows x 32 column-pairs
#pragma unroll 2
    for (int i = tid; i < 32 * 32; i += BLOCK_THREADS) {
      int pr = i >> 5, n2 = (i & 31) << 1;
      float2 ra = *(const float2*)&W2[(2 * pr + 0) * 64 + n2];
      float2 rb = *(const float2*)&W2[(2 * pr + 1) * 64 + n2];
      *(float4*)&lW2[pr * P_PITCH + 2 * n2] = make_float4(ra.x, rb.x, ra.y, rb.y);
    }
    if (tid < 64) {
      lW3[tid] = W3[tid];
      lb1[tid] = b1[tid];
      lb2[tid] = b2[tid];
    }
  }
  __syncthreads();

  const int node = blockIdx.x * WAVES_PER_BLOCK + wave;
  if (node >= N_NODES) return; // wave-uniform

  float* nbr = nodeBase + wave * NODE_F;
  float* h1 = nbr + DEGREE * R_PITCH;
  float* uArr = h1 + DEGREE * R_PITCH;
  float* att = uArr + EMBED;

  // ---- gather: target row + 32 neighbor rows -> LDS (all L2 hits: table is 25.6MB) ----
  {
    long uix = (long)nodes[node];
    const float4* urow = (const float4*)(emb + uix * EMBED);
    if (lane < 16) {
      float4 v = urow[lane];
      *(float4*)&uArr[lane * 4] = v;
    }
    long nix = (long)nbr_idx[node * DEGREE + lane]; // lane k owns neighbor k
    const float4* nrow = (const float4*)(emb + nix * EMBED);
    float* drow = nbr + lane * R_PITCH;
#pragma unroll
    for (int i = 0; i < 16; ++i) {
      float4 v = nrow[i];
      *(float4*)&drow[4 * i] = v;
    }
  }
  lds_fence();

  const int bcol = lo * 2; // column offset within a pair-row (2 floats per column)

  // ---- layer 1: h1 = relu([nbr | u] @ W1 + b1), f32 WMMA 16x16x4 ----
  const v8f vzero = {0.f, 0.f, 0.f, 0.f, 0.f, 0.f, 0.f, 0.f};
  v8f acc[2][4];
#pragma unroll
  for (int mt = 0; mt < 2; ++mt)
#pragma unroll
    for (int nt = 0; nt < 4; ++nt) acc[mt][nt] = vzero;

  // K = 0..63 : neighbor features (A rows vary)
  for (int kk = 0; kk < 16; ++kk) {
    const int k0 = kk * 4 + hi * 2;
    v2f a0 = *(const v2f*)&nbr[lo * R_PITCH + k0];
    v2f a1 = *(const v2f*)&nbr[(16 + lo) * R_PITCH + k0];
    const float* bpr = &lW1[(2 * kk + hi) * P_PITCH + bcol];
#pragma unroll
    for (int nt = 0; nt < 4; ++nt) {
      v2f b = *(const v2f*)&bpr[nt * 32]; // {W1[k0][n], W1[k0+1][n]} contiguous
      acc[0][nt] = wmma_f32(a0, b, acc[0][nt]);
      acc[1][nt] = wmma_f32(a1, b, acc[1][nt]);
    }
  }
  // K = 64..127 : broadcast target-user features (A identical across rows & M-tiles)
  for (int kk = 0; kk < 16; ++kk) {
    const int k0 = kk * 4 + hi * 2; // index into u
    v2f a = *(const v2f*)&uArr[k0];
    const float* bpr = &lW1[(32 + 2 * kk + hi) * P_PITCH + bcol]; // rows 64..127
#pragma unroll
    for (int nt = 0; nt < 4; ++nt) {
      v2f b = *(const v2f*)&bpr[nt * 32];
      acc[0][nt] = wmma_f32(a, b, acc[0][nt]);
      acc[1][nt] = wmma_f32(a, b, acc[1][nt]);
    }
  }

  // bias + relu, stage h1 (C-layout -> row-major LDS)
#pragma unroll
  for (int nt = 0; nt < 4; ++nt) {
    float bb = lb1[nt * 16 + lo];
#pragma unroll
    for (int mt = 0; mt < 2; ++mt) {
#pragma unroll
      for (int r = 0; r < 8; ++r) {
        float v = fmaxf(acc[mt][nt][r] + bb, 0.0f);
        h1[(mt * 16 + hi * 8 + r) * R_PITCH + nt * 16 + lo] = v;
      }
    }
  }
  lds_fence();

  // ---- layer 2: h2 = relu(h1 @ W2 + b2) (kept in accumulators) ----
  v8f acc2[2][4];
#pragma unroll
  for (int mt = 0; mt < 2; ++mt)
#pragma unroll
    for (int nt = 0; nt < 4; ++nt) acc2[mt][nt] = vzero;

  for (int kk = 0; kk < 16; ++kk) {
    const int k0 = kk * 4 + hi * 2;
    v2f a0 = *(const v2f*)&h1[lo * R_PITCH + k0];
    v2f a1 = *(const v2f*)&h1[(16 + lo) * R_PITCH + k0];
    const float* bpr = &lW2[(2 * kk + hi) * P_PITCH + bcol];
#pragma unroll
    for (int nt = 0; nt < 4; ++nt) {
      v2f b = *(const v2f*)&bpr[nt * 32];
      acc2[0][nt] = wmma_f32(a0, b, acc2[0][nt]);
      acc2[1][nt] = wmma_f32(a1, b, acc2[1][nt]);
    }
  }

  // ---- layer 3: scores = relu(h2) @ W3 (b3 omitted: softmax shift-invariant) ----
  float w3v[4], b2v[4];
#pragma unroll
  for (int nt = 0; nt < 4; ++nt) {
    w3v[nt] = lW3[nt * 16 + lo];
    b2v[nt] = lb2[nt * 16 + lo];
  }
#pragma unroll
  for (int mt = 0; mt < 2; ++mt) {
#pragma unroll
    for (int r = 0; r < 8; ++r) {
      float p = 0.f;
#pragma unroll
      for (int nt = 0; nt < 4; ++nt)
        p += fmaxf(acc2[mt][nt][r] + b2v[nt], 0.f) * w3v[nt];
      // reduce over the 16 N-lanes within each half-wave
      p += __shfl_xor(p, 1, 32);
      p += __shfl_xor(p, 2, 32);
      p += __shfl_xor(p, 4, 32);
      p += __shfl_xor(p, 8, 32);
      if (lo == 0) att[mt * 16 + hi * 8 + r] = p; // scores
    }
  }
  lds_fence();

  // ---- softmax over 32 neighbors (one score per lane) ----
  {
    float s = att[lane];
    float m = s;
    m = fmaxf(m, __shfl_xor(m, 16, 32));
    m = fmaxf(m, __shfl_xor(m, 8, 32));
    m = fmaxf(m, __shfl_xor(m, 4, 32));
    m = fmaxf(m, __shfl_xor(m, 2, 32));
    m = fmaxf(m, __shfl_xor(m, 1, 32));
    float e = __expf(s - m);
    float sum = e;
    sum += __shfl_xor(sum, 16, 32);
    sum += __shfl_xor(sum, 8, 32);
    sum += __shfl_xor(sum, 4, 32);
    sum += __shfl_xor(sum, 2, 32);
    sum += __shfl_xor(sum, 1, 32);
    att[lane] = e / sum;
  }
  lds_fence();

  // ---- weighted aggregation: out[node][:] = att^T @ nbr; lane owns dims 2l,2l+1 ----
  {
    v2f accO = {0.f, 0.f};
#pragma unroll
    for (int k = 0; k < DEGREE; ++k) {
      float w = att[k]; // broadcast
      v2f nb = *(const v2f*)&nbr[k * R_PITCH + 2 * lane];
      accO.x = fmaf(w, nb.x, accO.x);
      accO.y = fmaf(w, nb.y, accO.y);
    }
    *(v2f*)&out[(long)node * EMBED + 2 * lane] = accO;
  }
}

extern "C" void kernel_launch(void* const* d_in, const int* in_sizes, int n_in,
                              void* d_out, int out_size, void* d_ws, size_t ws_size,
                              hipStream_t stream) {
  (void)in_sizes; (void)n_in; (void)out_size; (void)d_ws; (void)ws_size;
  const int* nodes = (const int*)d_in[0];
  const int* nbr = (const int*)d_in[1];
  const float* emb = (const float*)d_in[2];
  const float* W1 = (const float*)d_in[3];
  const float* b1 = (const float*)d_in[4];
  const float* W2 = (const float*)d_in[5];
  const float* b2 = (const float*)d_in[6];
  const float* W3 = (const float*)d_in[7];
  // d_in[8] (b3) intentionally unused: softmax(x + c) == softmax(x)
  float* out = (float*)d_out;

  dim3 grid(N_NODES / WAVES_PER_BLOCK);
  dim3 block(BLOCK_THREADS);
  size_t smem = (size_t)SMEM_F * sizeof(float);
  hipLaunchKernelGGL(social_agg_kernel, grid, block, smem, stream,
                     nodes, nbr, emb, W1, b1, W2, b2, W3, out);
}